// GAT_TCN_89077621719520
// MI455X (gfx1250) — compile-verified
//
#include <hip/hip_runtime.h>
#include <hip/hip_bf16.h>

#define NN      2000
#define EE      128000
#define E2      130000      // EE + NN self loops
#define BB      8
#define HEADS   3
#define HID     64
#define HC      192         // HEADS*HID
#define T_IN    20
#define EPSV    1e-5f
#define NEGS    0.2f

typedef __attribute__((ext_vector_type(2))) float v2f;
typedef __attribute__((ext_vector_type(8))) float v8f;

static inline int cdiv(int a, int b) { return (a + b - 1) / b; }

// ---------------------------------------------------------------------------
// GAT conv kernels
// ---------------------------------------------------------------------------

__global__ void k_conv_init(float* __restrict__ den, float* __restrict__ out,
                            const float* __restrict__ bias) {
  int idx = blockIdx.x * blockDim.x + threadIdx.x;
  if (idx < NN * HEADS) den[idx] = 0.f;
  if (idx < NN * HC)    out[idx] = bias[idx % HC];
}

__global__ void k_wedot(const float* __restrict__ We, const float* __restrict__ ae,
                        float* __restrict__ wedot) {
  int h = threadIdx.x;
  if (h < HEADS) {
    float s = 0.f;
    for (int c = 0; c < HID; ++c) s += We[h * HID + c] * ae[h * HID + c];
    wedot[h] = s;
  }
}

// xl = x_in @ W   (x_in: [N, ind], W: [ind, HC])
__global__ void k_xl(const float* __restrict__ x, const float* __restrict__ W,
                     int ind, float* __restrict__ xl) {
  int idx = blockIdx.x * blockDim.x + threadIdx.x;
  if (idx >= NN * HC) return;
  int n = idx / HC, hc = idx % HC;
  float s = 0.f;
  for (int k = 0; k < ind; ++k) s += x[n * ind + k] * W[k * HC + hc];
  xl[idx] = s;
}

__global__ void k_attn(const float* __restrict__ xl, const float* __restrict__ as_,
                       const float* __restrict__ ad_, float* __restrict__ asrc,
                       float* __restrict__ adst) {
  int idx = blockIdx.x * blockDim.x + threadIdx.x;
  if (idx >= NN * HEADS) return;
  int n = idx / HEADS, h = idx % HEADS;
  float s = 0.f, d = 0.f;
  for (int c = 0; c < HID; ++c) {
    float v = xl[n * HC + h * HID + c];
    s += v * as_[h * HID + c];
    d += v * ad_[h * HID + c];
  }
  asrc[idx] = s;
  adst[idx] = d;
}

// per-(edge,head): alpha -> exp -> denominator (softmax is shift invariant)
__global__ void k_edge1(const int* __restrict__ ei, const float* __restrict__ ew,
                        const float* __restrict__ asrc, const float* __restrict__ adst,
                        const float* __restrict__ wedot, float* __restrict__ exbuf,
                        float* __restrict__ den) {
  int idx = blockIdx.x * blockDim.x + threadIdx.x;
  if (idx >= E2 * HEADS) return;
  int i = idx / HEADS, h = idx % HEADS;
  int src, dst; float w;
  if (i < EE) { src = ei[i]; dst = ei[EE + i]; w = ew[i]; }
  else        { src = i - EE; dst = i - EE;    w = 0.f;  }
  float alpha = asrc[src * HEADS + h] + adst[dst * HEADS + h] + w * wedot[h];
  alpha = alpha > 0.f ? alpha : NEGS * alpha;           // leaky relu
  float ex = __expf(alpha);
  exbuf[idx] = ex;
  atomicAdd(&den[dst * HEADS + h], ex);
}

// per-(edge,feature): weighted scatter-add into destination
__global__ void k_edge2(const int* __restrict__ ei, const float* __restrict__ exbuf,
                        const float* __restrict__ den, const float* __restrict__ xl,
                        float* __restrict__ out) {
  int idx = blockIdx.x * blockDim.x + threadIdx.x;
  if (idx >= E2 * HC) return;
  int i = idx / HC, hc = idx % HC, h = hc >> 6;
  int src, dst;
  if (i < EE) { src = ei[i]; dst = ei[EE + i]; }
  else        { src = i - EE; dst = i - EE;    }
  float attn = exbuf[i * HEADS + h] / (den[dst * HEADS + h] + 1e-16f);
  atomicAdd(&out[dst * HC + hc], attn * xl[src * HC + hc]);
}

__global__ void k_bnrelu(const float* __restrict__ in, const float* __restrict__ g,
                         const float* __restrict__ be, const float* __restrict__ mu,
                         const float* __restrict__ var, float* __restrict__ out) {
  int idx = blockIdx.x * blockDim.x + threadIdx.x;
  if (idx >= NN * HC) return;
  int c = idx % HC;
  float v = (in[idx] - mu[c]) * rsqrtf(var[c] + EPSV) * g[c] + be[c];
  out[idx] = v > 0.f ? v : 0.f;
}

__global__ void k_add_bnrelu(const float* __restrict__ in, const float* __restrict__ res,
                             const float* __restrict__ g, const float* __restrict__ be,
                             const float* __restrict__ mu, const float* __restrict__ var,
                             float* __restrict__ out) {
  int idx = blockIdx.x * blockDim.x + threadIdx.x;
  if (idx >= NN * HC) return;
  int c = idx % HC;
  float h = in[idx] + res[idx];
  float v = (h - mu[c]) * rsqrtf(var[c] + EPSV) * g[c] + be[c];
  out[idx] = v > 0.f ? v : 0.f;
}

__global__ void k_branchfc(const float* __restrict__ h, const float* __restrict__ W,
                           const float* __restrict__ b, float* __restrict__ out) {
  int idx = blockIdx.x * blockDim.x + threadIdx.x;
  if (idx >= NN * HID) return;
  int n = idx / HID, c = idx % HID;
  float s = b[c];
  for (int hc = 0; hc < HC; ++hc) s += h[n * HC + hc] * W[hc * HID + c];
  out[idx] = s;
}

// TS[n*64+c] = t.T.reshape + s.T.reshape quirk: flat i -> t[i%2000, i/2000]
__global__ void k_ts(const float* __restrict__ tb, const float* __restrict__ sb,
                     float* __restrict__ TS) {
  int i = blockIdx.x * blockDim.x + threadIdx.x;
  if (i >= NN * HID) return;
  int row = i % NN, col = i / NN;
  TS[i] = tb[row * HID + col] + sb[row * HID + col];
}

// ---------------------------------------------------------------------------
// Fused tcn1 + fusions + tcn2 + BatchNorm(node) -> res_flat [16][80000]
// rows b>=8 are zero padding for the WMMA A matrix.
// ---------------------------------------------------------------------------
__global__ void k_fused_res(
    const float* __restrict__ x, const float* __restrict__ r,
    const float* __restrict__ w1a, const float* __restrict__ w2a,
    const float* __restrict__ w3a, const float* __restrict__ b1a,
    const float* __restrict__ b2a, const float* __restrict__ b3a,
    const float* __restrict__ w1b, const float* __restrict__ w2b,
    const float* __restrict__ w3b, const float* __restrict__ b1b,
    const float* __restrict__ b2b, const float* __restrict__ b3b,
    const float* __restrict__ lrW, const float* __restrict__ lrb,
    const float* __restrict__ TS,
    const float* __restrict__ bng, const float* __restrict__ bnb,
    const float* __restrict__ bnm, const float* __restrict__ bnv,
    float* __restrict__ res) {
  int idx = blockIdx.x * blockDim.x + threadIdx.x;
  if (idx >= 16 * 40 * NN) return;
  int b = idx / (40 * NN);
  int rem = idx % (40 * NN);
  int t = rem / NN;
  int n = rem % NN;
  if (b >= BB) { res[idx] = 0.f; return; }

  float P = b1b[0], Q = b2b[0], R = b3b[0];
  for (int k = 0; k < 3; ++k) {
    int tc = t + k - 1;
    if (tc < 0 || tc >= 40) continue;
    if (tc < T_IN) {
      // xt channel vector from gated tcn1 of x, recomputed on the fly
      float xm = (tc - 1 >= 0)   ? x[(b * T_IN + tc - 1) * NN + n] : 0.f;
      float xc =                   x[(b * T_IN + tc)     * NN + n];
      float xp = (tc + 1 < T_IN) ? x[(b * T_IN + tc + 1) * NN + n] : 0.f;
      for (int c = 0; c < HID; ++c) {
        float p1 = w1a[c * 3 + 0] * xm + w1a[c * 3 + 1] * xc + w1a[c * 3 + 2] * xp + b1a[c];
        float q1 = w2a[c * 3 + 0] * xm + w2a[c * 3 + 1] * xc + w2a[c * 3 + 2] * xp + b2a[c];
        float r1 = w3a[c * 3 + 0] * xm + w3a[c * 3 + 1] * xc + w3a[c * 3 + 2] * xp + b3a[c];
        float h1 = p1 * (1.f / (1.f + __expf(-q1))) + r1;
        float cin = h1 > 0.f ? h1 : 0.f;
        P += w1b[c * 3 + k] * cin;
        Q += w2b[c * 3 + k] * cin;
        R += w3b[c * 3 + k] * cin;
      }
    } else {
      int tt = tc - T_IN;
      float rv = r[(b * T_IN + tt) * NN + n];
      for (int c = 0; c < HID; ++c) {
        float f = rv * lrW[c] + lrb[c];
        f = f > 0.f ? f : 0.f;                         // relu(linear(r))
        float cin = f + TS[n * HID + c];               // + t_r + s_r broadcast
        P += w1b[c * 3 + k] * cin;
        Q += w2b[c * 3 + k] * cin;
        R += w3b[c * 3 + k] * cin;
      }
    }
  }
  float H = P * (1.f / (1.f + __expf(-Q))) + R;
  H = H > 0.f ? H : 0.f;
  float o = (H - bnm[n]) * rsqrtf(bnv[n] + EPSV) * bng[n] + bnb[n];
  res[idx] = o;
}

// ---------------------------------------------------------------------------
// WMMA f32 16x16x4 GEMMs (wave32, one wave per block)
// A: [16][K] row-major (rows >=8 are zero)   B: [K][Ncols] row-major
// A frag: lane l -> M = l&15 ; V0/V1 hold K = kr, kr+1 with kr = (l<16)?0:2
// B frag: V0/V1 hold rows kr, kr+1 at column j0 + (l&15)
// C/D  : lanes 0-15, vgpr v -> (M=v, N=lane)
// ---------------------------------------------------------------------------
#define KCH1 1600   // 80000 / 50
#define KCH2 256    // 2048 / 8

__global__ void k_gemm1(const float* __restrict__ A, const float* __restrict__ W,
                        float* __restrict__ part) {
  int jt = blockIdx.x;           // 0..127 -> 2048 cols
  int kc = blockIdx.y;           // 0..49
  int lane = threadIdx.x;        // 0..31
  int m = lane & 15;
  int kr = (lane < 16) ? 0 : 2;
  int j0 = jt * 16;
  v8f acc = {};
  const float* Ap = A + (size_t)m * 80000 + kc * KCH1 + kr;
  const float* Wp = W + (size_t)(kc * KCH1 + kr) * 2048 + j0 + m;
  for (int k = 0; k < KCH1; k += 4) {
    v2f a, bf;
    a.x = Ap[0]; a.y = Ap[1];
    bf.x = Wp[0]; bf.y = Wp[2048];
    Ap += 4;
    Wp += (size_t)4 * 2048;
    acc = __builtin_amdgcn_wmma_f32_16x16x4_f32(false, a, false, bf,
                                                (short)0, acc, false, false);
  }
  if (lane < 16) {
#pragma unroll
    for (int v = 0; v < 8; ++v)
      part[(size_t)(kc * 8 + v) * 2048 + j0 + lane] = acc[v];
  }
}

__global__ void k_reduce1(const float* __restrict__ part, const float* __restrict__ fcb,
                          float* __restrict__ acc1) {
  int idx = blockIdx.x * blockDim.x + threadIdx.x;
  if (idx >= 16 * 2048) return;
  int b = idx >> 11, j = idx & 2047;
  float s = 0.f;
  if (b < BB) {
    s = fcb[j];
    for (int kc = 0; kc < 50; ++kc) s += part[(size_t)(kc * 8 + b) * 2048 + j];
  }
  acc1[idx] = s;     // pre-relu; rows >=8 zero (relu(0)=0 feeds GEMM2 cleanly)
}

__global__ void k_gemm2(const float* __restrict__ A, const float* __restrict__ W,
                        float* __restrict__ part) {
  int jt = blockIdx.x;           // 0..124 -> 2000 cols
  int kc = blockIdx.y;           // 0..7
  int lane = threadIdx.x;
  int m = lane & 15;
  int kr = (lane < 16) ? 0 : 2;
  int j0 = jt * 16;
  v8f acc = {};
  const float* Ap = A + (size_t)m * 2048 + kc * KCH2 + kr;
  const float* Wp = W + (size_t)(kc * KCH2 + kr) * 2000 + j0 + m;
  for (int k = 0; k < KCH2; k += 4) {
    v2f a, bf;
    a.x = fmaxf(Ap[0], 0.f);     // relu applied on the fly
    a.y = fmaxf(Ap[1], 0.f);
    bf.x = Wp[0]; bf.y = Wp[2000];
    Ap += 4;
    Wp += (size_t)4 * 2000;
    acc = __builtin_amdgcn_wmma_f32_16x16x4_f32(false, a, false, bf,
                                                (short)0, acc, false, false);
  }
  if (lane < 16) {
#pragma unroll
    for (int v = 0; v < 8; ++v)
      part[(size_t)(kc * 8 + v) * 2000 + j0 + lane] = acc[v];
  }
}

__global__ void k_reduce2(const float* __restrict__ part, const float* __restrict__ fc1b,
                          float* __restrict__ out) {
  int idx = blockIdx.x * blockDim.x + threadIdx.x;
  if (idx >= BB * NN) return;
  int b = idx / NN, mcol = idx % NN;
  float s = fc1b[mcol];
  for (int kc = 0; kc < 8; ++kc) s += part[(size_t)(kc * 8 + b) * 2000 + mcol];
  out[idx] = s;
}

// ---------------------------------------------------------------------------
// Host orchestration
// ---------------------------------------------------------------------------
namespace {

struct Scratch {
  float *xl, *asrc, *adst, *wedot, *exbuf, *den;
  float *convA, *x0, *h1;
  float *tout, *sout, *TS;
  float *resf, *part1, *acc1, *part2;
};

void run_conv(hipStream_t s, const float* x_in, int ind, const float* W,
              const float* as_, const float* ad_, const float* We, const float* ae,
              const float* bias, const int* ei, const float* ew, const Scratch& w,
              float* out) {
  k_conv_init<<<cdiv(NN * HC, 256), 256, 0, s>>>(w.den, out, bias);
  k_wedot<<<1, 32, 0, s>>>(We, ae, w.wedot);
  k_xl<<<cdiv(NN * HC, 256), 256, 0, s>>>(x_in, W, ind, w.xl);
  k_attn<<<cdiv(NN * HEADS, 256), 256, 0, s>>>(w.xl, as_, ad_, w.asrc, w.adst);
  k_edge1<<<cdiv(E2 * HEADS, 256), 256, 0, s>>>(ei, ew, w.asrc, w.adst, w.wedot,
                                                w.exbuf, w.den);
  k_edge2<<<cdiv(E2 * HC, 256), 256, 0, s>>>(ei, w.exbuf, w.den, w.xl, out);
}

void run_branch(hipStream_t s, const float* feat, void* const* din, int G,
                const int* ei, const float* ew, const Scratch& w, float* out64) {
  auto F = [&](int i) { return (const float*)din[i]; };
  // conv0 -> bn0 -> relu -> x0
  run_conv(s, feat, T_IN, F(G + 0), F(G + 1), F(G + 2), F(G + 3), F(G + 4), F(G + 5),
           ei, ew, w, w.convA);
  k_bnrelu<<<cdiv(NN * HC, 256), 256, 0, s>>>(w.convA, F(G + 6), F(G + 7), F(G + 8),
                                              F(G + 9), w.x0);
  // layers[0] -> bns[0] -> relu -> h1
  run_conv(s, feat, T_IN, F(G + 10), F(G + 11), F(G + 12), F(G + 13), F(G + 14),
           F(G + 15), ei, ew, w, w.convA);
  k_bnrelu<<<cdiv(NN * HC, 256), 256, 0, s>>>(w.convA, F(G + 22), F(G + 23), F(G + 24),
                                              F(G + 25), w.h1);
  // layers[1](h1) + x0 -> bns[1] -> relu -> h1
  run_conv(s, w.h1, HC, F(G + 16), F(G + 17), F(G + 18), F(G + 19), F(G + 20),
           F(G + 21), ei, ew, w, w.convA);
  k_add_bnrelu<<<cdiv(NN * HC, 256), 256, 0, s>>>(w.convA, w.x0, F(G + 26), F(G + 27),
                                                  F(G + 28), F(G + 29), w.h1);
  // fc head -> [N, HID]
  k_branchfc<<<cdiv(NN * HID, 256), 256, 0, s>>>(w.h1, F(G + 30), F(G + 31), out64);
}

} // namespace

extern "C" void kernel_launch(void* const* d_in, const int* in_sizes, int n_in,
                              void* d_out, int out_size, void* d_ws, size_t ws_size,
                              hipStream_t stream) {
  (void)in_sizes; (void)n_in; (void)out_size; (void)ws_size;
  const float* t_feat = (const float*)d_in[0];
  const float* s_feat = (const float*)d_in[1];
  const float* r      = (const float*)d_in[2];
  const float* x      = (const float*)d_in[3];
  const float* ew     = (const float*)d_in[4];
  const int*   ei     = (const int*)d_in[5];
  auto F = [&](int i) { return (const float*)d_in[i]; };

  // carve workspace (floats)
  float* w = (float*)d_ws;
  Scratch sc;
  sc.xl    = w; w += NN * HC;          // 384000
  sc.asrc  = w; w += NN * HEADS;
  sc.adst  = w; w += NN * HEADS;
  sc.wedot = w; w += 32;
  sc.exbuf = w; w += E2 * HEADS;       // 390000
  sc.den   = w; w += NN * HEADS;
  sc.convA = w; w += NN * HC;
  sc.x0    = w; w += NN * HC;
  sc.h1    = w; w += NN * HC;
  sc.tout  = w; w += NN * HID;
  sc.sout  = w; w += NN * HID;
  sc.TS    = w; w += NN * HID;
  sc.resf  = w; w += 16 * 80000;       // padded A matrix for GEMM1
  sc.part1 = w; w += 50 * 8 * 2048;
  sc.acc1  = w; w += 16 * 2048;        // padded A matrix for GEMM2
  sc.part2 = w; w += 8 * 8 * 2000;

  // GAT branches (gat1 params start at 6, gat2 at 38)
  run_branch(stream, t_feat, d_in, 6,  ei, ew, sc, sc.tout);
  run_branch(stream, s_feat, d_in, 38, ei, ew, sc, sc.sout);

  // transpose-reshape quirk fusion table
  k_ts<<<cdiv(NN * HID, 256), 256, 0, stream>>>(sc.tout, sc.sout, sc.TS);

  // fused tcn1 + fusions + tcn2 + node-BN -> res_flat [16][80000]
  k_fused_res<<<cdiv(16 * 40 * NN, 256), 256, 0, stream>>>(
      x, r,
      F(70), F(71), F(72), F(73), F(74), F(75),     // tcn1 w1..w3, b1..b3
      F(76), F(77), F(78), F(79), F(80), F(81),     // tcn2 w1..w3, b1..b3
      F(82), F(83),                                 // lr_W, lr_b
      sc.TS,
      F(84), F(85), F(86), F(87),                   // bn gamma/beta/mean/var
      sc.resf);

  // fc: [16,80000] x [80000,2048] via WMMA f32 16x16x4, K split into 50 chunks
  dim3 g1(128, 50);
  k_gemm1<<<g1, 32, 0, stream>>>(sc.resf, F(88), sc.part1);
  k_reduce1<<<cdiv(16 * 2048, 256), 256, 0, stream>>>(sc.part1, F(89), sc.acc1);

  // fc1: relu(acc1) x [2048,2000], K split into 8 chunks
  dim3 g2(125, 8);
  k_gemm2<<<g2, 32, 0, stream>>>(sc.acc1, F(90), sc.part2);
  k_reduce2<<<cdiv(BB * NN, 256), 256, 0, stream>>>(sc.part2, F(91), (float*)d_out);
}